// MeanVFE_59407987638869
// MI455X (gfx1250) — compile-verified
//
#include <hip/hip_runtime.h>
#include <stdint.h>

#ifndef __has_builtin
#define __has_builtin(x) 0
#endif

#define TPB 256
#define MAX_VOXK 160000
#define MAX_PTSK 32
#define GX 432
#define GY 496
#define GZ 1
#define TOTALVOX (GX * GY * GZ)            /* 214272 */
#define SCAN_ITEMS 4
#define SCAN_CHUNK (TPB * SCAN_ITEMS)      /* 1024 */
#define NSCAN_BLOCKS ((TOTALVOX + SCAN_CHUNK - 1) / SCAN_CHUNK) /* 210 */

// ---------------- CDNA5 async global->LDS plumbing ----------------
#if __has_builtin(__builtin_amdgcn_global_load_async_to_lds_b32)
#define USE_ASYNC_LDS 1
typedef __attribute__((address_space(1))) int glob_i32;
typedef __attribute__((address_space(3))) int lds_i32;
#else
#define USE_ASYNC_LDS 0
#endif

#if USE_ASYNC_LDS && __has_builtin(__builtin_amdgcn_global_load_async_to_lds_b128)
#define HAVE_ASYNC_B128 1
#define FAST_OPS 1   /* async instructions per wave per fast tile */
typedef int v4i __attribute__((ext_vector_type(4)));
typedef __attribute__((address_space(1))) v4i glob_v4i;
typedef __attribute__((address_space(3))) v4i lds_v4i;
#else
#define HAVE_ASYNC_B128 0
#define FAST_OPS 4
#endif

#if __has_builtin(__builtin_amdgcn_s_wait_asynccnt)
#define WAIT_ASYNC(n) __builtin_amdgcn_s_wait_asynccnt(n)
#elif defined(__AMDGCN__)
#define WAIT_ASYNC(n) asm volatile("s_wait_asynccnt %0" ::"i"(n))
#else
#define WAIT_ASYNC(n) ((void)0)
#endif

// ---------------- helpers ----------------
__device__ __forceinline__ void voxel_coords(float x, float y, float z,
                                             int& cx, int& cy, int& cz) {
  // Matches jnp.floor((xyz - low) / vsz) in f32 (true division, floorf).
  cx = (int)floorf((x - 0.0f)  / 0.16f);
  cy = (int)floorf((y + 39.68f) / 0.16f);
  cz = (int)floorf((z + 3.0f)  / 4.0f);
}

__global__ void k_fill_u32(uint32_t* __restrict__ p, int n) {
  int i = blockIdx.x * blockDim.x + threadIdx.x;
  if (i < n) p[i] = 0u;
}

// Pass A: occupancy bitmap (word per voxel; same-value races are benign).
__global__ void k_mark(const float* __restrict__ clouds, uint32_t* __restrict__ occ,
                       int N, int P) {
  int p = blockIdx.x * blockDim.x + threadIdx.x;
  if (p >= P) return;
  int b = p / N, n = p - b * N;
  const float* base = clouds + (size_t)b * 4u * (size_t)N + n;
  float x = base[0];
  float y = base[(size_t)N];
  float z = base[(size_t)2 * N];
  int cx, cy, cz;
  voxel_coords(x, y, z, cx, cy, cz);
  if (cx >= 0 && cx < GX && cy >= 0 && cy < GY && cz >= 0 && cz < GZ) {
    int lin = (cz * GY + cy) * GX + cx;
    occ[lin] = 1u;
  }
}

// Pass B1: per-1024-voxel exclusive scan + block totals.
__global__ void k_scan_local(const uint32_t* __restrict__ occ,
                             uint32_t* __restrict__ slotLocal,
                             uint32_t* __restrict__ blockSums) {
  __shared__ uint32_t sh[TPB];
  int tid = threadIdx.x;
  int base = blockIdx.x * SCAN_CHUNK + tid * SCAN_ITEMS;
  uint32_t v[SCAN_ITEMS];
  uint32_t tsum = 0;
#pragma unroll
  for (int i = 0; i < SCAN_ITEMS; ++i) {
    v[i] = (base + i < TOTALVOX) ? occ[base + i] : 0u;
    tsum += v[i];
  }
  sh[tid] = tsum;
  __syncthreads();
#pragma unroll
  for (int off = 1; off < TPB; off <<= 1) {
    uint32_t t = (tid >= off) ? sh[tid - off] : 0u;
    __syncthreads();
    sh[tid] += t;
    __syncthreads();
  }
  uint32_t run = sh[tid] - tsum;  // exclusive prefix of this thread's chunk
#pragma unroll
  for (int i = 0; i < SCAN_ITEMS; ++i) {
    if (base + i < TOTALVOX) slotLocal[base + i] = run;
    run += v[i];
  }
  if (tid == TPB - 1) blockSums[blockIdx.x] = sh[tid];
}

// Pass B2: serial exclusive scan over 210 block sums (trivial work).
__global__ void k_scan_blocks(uint32_t* __restrict__ bs, int nb) {
  if (blockIdx.x == 0 && threadIdx.x == 0) {
    uint32_t run = 0;
    for (int i = 0; i < nb; ++i) {
      uint32_t t = bs[i];
      bs[i] = run;
      run += t;
    }
  }
}

// A tile is "slow" if it runs past P or straddles a batch-plane boundary.
// Uniform per tile -> uniform per wave -> ASYNCcnt bookkeeping stays exact.
__device__ __forceinline__ bool tile_slow(int tile, int N, int P) {
  int s = tile * TPB;
  int e = s + TPB - 1;
  return (e >= P) || (s / N != e / N);
}

// Stage one 256-point tile (4 channel planes) into LDS: dst[c][i] = ch c of pt i.
__device__ __forceinline__ void issue_tile(const float* __restrict__ clouds,
                                           float (*dst)[TPB],
                                           int tile, int tid, int N, int P,
                                           bool slow) {
#if HAVE_ASYNC_B128
  if (!slow) {
    // Wide path: lane (c,g) DMAs 4 consecutive points of channel c (16B, aligned).
    int c = tid & 3;
    int g = tid >> 2;
    int p0 = tile * TPB + g * 4;
    int b = p0 / N, n0 = p0 - b * N;
    const float* gp = clouds + ((size_t)b * 4u + (size_t)c) * (size_t)N + n0;
    __builtin_amdgcn_global_load_async_to_lds_b128(
        (glob_v4i*)(uintptr_t)gp,
        (lds_v4i*)(uint32_t)(uintptr_t)&dst[c][g * 4],
        /*offset=*/0, /*cpol=*/0);
    return;
  }
#else
  (void)slow;
#endif
  int p = tile * TPB + tid;
  if (p >= P) p = P - 1;  // clamp; masked later at compute
  int b = p / N, n = p - b * N;
  const float* g = clouds + (size_t)b * 4u * (size_t)N + n;
#if USE_ASYNC_LDS
#pragma unroll
  for (int c = 0; c < 4; ++c) {
    __builtin_amdgcn_global_load_async_to_lds_b32(
        (glob_i32*)(uintptr_t)(g + (size_t)c * N),
        (lds_i32*)(uint32_t)(uintptr_t)&dst[c][tid],
        /*offset=*/0, /*cpol=*/0);
  }
#else
#pragma unroll
  for (int c = 0; c < 4; ++c) dst[c][tid] = g[(size_t)c * N];
#endif
}

// Pass C: double-buffered async point stream -> L2-resident scatter-add.
__global__ void k_accum(const float* __restrict__ clouds,
                        const uint32_t* __restrict__ slotLocal,
                        const uint32_t* __restrict__ blockOff,
                        int* __restrict__ counts,
                        float* __restrict__ out,
                        int N, int P) {
  __shared__ float sh[2][4][TPB];  // 8 KB of the 320 KB/WGP pool
  const int tid = threadIdx.x;
  const int nTiles = (P + TPB - 1) / TPB;
  const int stride = (int)gridDim.x;
  int tile = (int)blockIdx.x;
  int buf = 0;
  if (tile < nTiles) issue_tile(clouds, sh[0], tile, tid, N, P, tile_slow(tile, N, P));
  for (; tile < nTiles; tile += stride, buf ^= 1) {
    int nxt = tile + stride;
    if (nxt < nTiles) {
      bool slowNext = tile_slow(nxt, N, P);
      issue_tile(clouds, sh[buf ^ 1], nxt, tid, N, P, slowNext);
      // In-order completion: wait until only the next tile's ops are outstanding.
      if (slowNext) WAIT_ASYNC(4);
      else          WAIT_ASYNC(FAST_OPS);
    } else {
      WAIT_ASYNC(0);
    }
    float x = sh[buf][0][tid];
    float y = sh[buf][1][tid];
    float z = sh[buf][2][tid];
    float w = sh[buf][3][tid];
    int p = tile * TPB + tid;
    if (p < P) {
      int cx, cy, cz;
      voxel_coords(x, y, z, cx, cy, cz);
      if (cx >= 0 && cx < GX && cy >= 0 && cy < GY && cz >= 0 && cz < GZ) {
        int lin = (cz * GY + cy) * GX + cx;
        uint32_t slot = slotLocal[lin] + blockOff[lin >> 10];
        if (slot < (uint32_t)MAX_VOXK) {
          int r = atomicAdd(&counts[slot], 1);
          if (r < MAX_PTSK) {  // cap: first 32 points per voxel
            float* o = out + (size_t)slot * 4u;
            atomicAdd(o + 0, x);
            atomicAdd(o + 1, y);
            atomicAdd(o + 2, z);
            atomicAdd(o + 3, w);
          }
        }
      }
    }
  }
}

// Pass D: mean = sum / max(min(cnt,32),1)
__global__ void k_final(float* __restrict__ out, const int* __restrict__ counts) {
  int v = blockIdx.x * blockDim.x + threadIdx.x;
  if (v >= MAX_VOXK) return;
  int c = counts[v];
  if (c > MAX_PTSK) c = MAX_PTSK;
  float norm = (c > 1) ? (float)c : 1.0f;
  float4* o4 = (float4*)out + v;
  float4 s = *o4;
  s.x /= norm; s.y /= norm; s.z /= norm; s.w /= norm;
  *o4 = s;
}

extern "C" void kernel_launch(void* const* d_in, const int* in_sizes, int n_in,
                              void* d_out, int out_size, void* d_ws, size_t ws_size,
                              hipStream_t stream) {
  (void)n_in; (void)out_size; (void)ws_size;
  const float* clouds = (const float*)d_in[0];
  const int BCN = in_sizes[0];      // B*C*N = 16*N
  const int N = BCN / 16;
  const int P = 4 * N;
  float* out = (float*)d_out;

  // Workspace layout (~2.4 MB)
  uint32_t* occ       = (uint32_t*)d_ws;            // TOTALVOX words
  uint32_t* slotLocal = occ + TOTALVOX;             // TOTALVOX words
  uint32_t* blockOff  = slotLocal + TOTALVOX;       // 256 words (210 used)
  int*      counts    = (int*)(blockOff + 256);     // MAX_VOXK words

  int nb;
  nb = (TOTALVOX + TPB - 1) / TPB;
  k_fill_u32<<<nb, TPB, 0, stream>>>(occ, TOTALVOX);
  nb = (MAX_VOXK + TPB - 1) / TPB;
  k_fill_u32<<<nb, TPB, 0, stream>>>((uint32_t*)counts, MAX_VOXK);
  nb = (MAX_VOXK * 4 + TPB - 1) / TPB;
  k_fill_u32<<<nb, TPB, 0, stream>>>((uint32_t*)out, MAX_VOXK * 4);

  nb = (P + TPB - 1) / TPB;
  k_mark<<<nb, TPB, 0, stream>>>(clouds, occ, N, P);

  k_scan_local<<<NSCAN_BLOCKS, TPB, 0, stream>>>(occ, slotLocal, blockOff);
  k_scan_blocks<<<1, 32, 0, stream>>>(blockOff, NSCAN_BLOCKS);

  int nTiles = (P + TPB - 1) / TPB;
  int accBlocks = nTiles < 1024 ? nTiles : 1024;  // grid-stride -> real double buffering
  k_accum<<<accBlocks, TPB, 0, stream>>>(clouds, slotLocal, blockOff, counts, out, N, P);

  nb = (MAX_VOXK + TPB - 1) / TPB;
  k_final<<<nb, TPB, 0, stream>>>(out, counts);
}